// Attention_1_59227599012036
// MI455X (gfx1250) — compile-verified
//
#include <hip/hip_runtime.h>
#include <hip/hip_bf16.h>
#include <math.h>

#define B_ 32
#define T_ 2048
#define H_ 1024
#define DOUT_ 128
#define NSEG_ 16
#define TSEG_ (T_ / NSEG_)   // 128

typedef __attribute__((ext_vector_type(2))) float v2f;
typedef __attribute__((ext_vector_type(8))) float v8f;

// ---------------------------------------------------------------------------
// Kernel 1: v[b,k] = sum_h h_t[b,h] * W_score[h,k]
//   A = h_t rows of hidden (row stride T*H), 32x1024 ; B = W_score 1024x1024
//   WMMA f32 16x16x4. 2 M-tiles x 64 N-tiles = 128 waves.
// ---------------------------------------------------------------------------
__global__ void k_scorevec_wmma(const float* __restrict__ hidden,
                                const float* __restrict__ Wscore,
                                float* __restrict__ v) {
    const unsigned lane = threadIdx.x & 31u;
    const unsigned wid  = (blockIdx.x * blockDim.x + threadIdx.x) >> 5; // 0..127
    const int m0 = (int)(wid >> 6) << 4;   // 0 or 16
    const int n0 = (int)(wid & 63u) << 4;  // 0..1008

    const float* A = hidden + (size_t)(T_ - 1) * H_;  // h_t, row stride T*H
    const size_t lda = (size_t)T_ * H_;

    const int mrow = m0 + (int)(lane & 15u);
    const int ncol = n0 + (int)(lane & 15u);
    const int kh   = ((int)(lane >> 4)) << 1;  // 0 or 2 (K half per lane group)

    v8f acc = {};
    for (int k = 0; k < H_; k += 4) {
        v2f a, b;
        a[0] = A[(size_t)mrow * lda + (k + kh)];
        a[1] = A[(size_t)mrow * lda + (k + kh + 1)];
        b[0] = Wscore[(size_t)(k + kh)     * H_ + ncol];
        b[1] = Wscore[(size_t)(k + kh + 1) * H_ + ncol];
        acc = __builtin_amdgcn_wmma_f32_16x16x4_f32(
                  false, a, false, b, (short)0, acc, false, false);
    }
    const int n  = n0 + (int)(lane & 15u);
    const int mb = m0 + (((int)(lane >> 4)) << 3);
    for (int r = 0; r < 8; ++r)
        v[(size_t)(mb + r) * H_ + n] = acc[r];
}

// ---------------------------------------------------------------------------
// Kernel 2: score[b,t] = dot(hidden[b,t,:], v[b,:])   (memory-bound pass #1)
//   One wave per (b,t); float4 (b128) coalesced loads; shuffle reduction.
// ---------------------------------------------------------------------------
__global__ void k_score(const float* __restrict__ hidden,
                        const float* __restrict__ v,
                        float* __restrict__ score) {
    const unsigned lane = threadIdx.x & 31u;
    const unsigned w    = (blockIdx.x * blockDim.x + threadIdx.x) >> 5;
    const int b = (int)(w >> 11);        // / T_
    const int t = (int)(w & (T_ - 1));
    const float4* hrow = (const float4*)(hidden + ((size_t)b * T_ + t) * H_);
    const float4* vrow = (const float4*)(v + (size_t)b * H_);

    float acc = 0.f;
    for (int i = (int)lane; i < H_ / 4; i += 32) {
        float4 a = hrow[i];
        float4 c = vrow[i];
        acc = fmaf(a.x, c.x, acc);
        acc = fmaf(a.y, c.y, acc);
        acc = fmaf(a.z, c.z, acc);
        acc = fmaf(a.w, c.w, acc);
    }
    for (int off = 16; off; off >>= 1)
        acc += __shfl_xor(acc, off, 32);
    if (lane == 0)
        score[(size_t)b * T_ + t] = acc;
}

// ---------------------------------------------------------------------------
// Kernel 3: softmax over t for each b (in-place on score). 1 block per b.
// ---------------------------------------------------------------------------
__global__ void k_softmax(float* __restrict__ score) {
    __shared__ float red[256];
    const int b   = blockIdx.x;
    const int tid = threadIdx.x;
    float* s = score + (size_t)b * T_;

    float m = -INFINITY;
    for (int t = tid; t < T_; t += 256) m = fmaxf(m, s[t]);
    red[tid] = m; __syncthreads();
    for (int o = 128; o; o >>= 1) {
        if (tid < o) red[tid] = fmaxf(red[tid], red[tid + o]);
        __syncthreads();
    }
    m = red[0]; __syncthreads();

    float sum = 0.f;
    for (int t = tid; t < T_; t += 256) sum += __expf(s[t] - m);
    red[tid] = sum; __syncthreads();
    for (int o = 128; o; o >>= 1) {
        if (tid < o) red[tid] += red[tid + o];
        __syncthreads();
    }
    const float inv = 1.0f / red[0];
    for (int t = tid; t < T_; t += 256) s[t] = __expf(s[t] - m) * inv;
}

// ---------------------------------------------------------------------------
// Kernel 4a: partial[seg][b][h] = sum_{t in seg} w[b,t]*hidden[b,t,h]
//   Grid: B*NSEG blocks x 256 threads. Each thread owns 4 contiguous h
//   (float4 -> global_load_b128, fully coalesced). Weights staged in LDS.
// ---------------------------------------------------------------------------
__global__ void k_context_part(const float* __restrict__ hidden,
                               const float* __restrict__ w,
                               float* __restrict__ partial) {
    __shared__ float ws[TSEG_];
    const int b   = (int)(blockIdx.x >> 4);      // / NSEG_
    const int seg = (int)(blockIdx.x & (NSEG_ - 1));
    const int t0  = seg * TSEG_;

    const float4* hb = (const float4*)(hidden + (size_t)b * T_ * H_);
    const float*  wb = w + (size_t)b * T_ + t0;
    if (threadIdx.x < TSEG_) ws[threadIdx.x] = wb[threadIdx.x];
    __syncthreads();

    const int h4 = (int)threadIdx.x;             // float4 index, covers H
    float4 acc = make_float4(0.f, 0.f, 0.f, 0.f);
    for (int tt = 0; tt < TSEG_; ++tt) {
        const float wt = ws[tt];
        const float4 hv = hb[(size_t)(t0 + tt) * (H_ / 4) + h4];
        acc.x = fmaf(wt, hv.x, acc.x);
        acc.y = fmaf(wt, hv.y, acc.y);
        acc.z = fmaf(wt, hv.z, acc.z);
        acc.w = fmaf(wt, hv.w, acc.w);
    }
    float4* p4 = (float4*)(partial + ((size_t)seg * B_ + b) * H_);
    p4[h4] = acc;
}

// ---------------------------------------------------------------------------
// Kernel 4b: reduce segments -> P[b] = [context[b,:], h_t[b,:]]
// ---------------------------------------------------------------------------
__global__ void k_context_reduce(const float* __restrict__ partial,
                                 const float* __restrict__ hidden,
                                 float* __restrict__ P) {
    const int b  = blockIdx.x;
    const int h4 = (int)threadIdx.x;
    float4 acc = make_float4(0.f, 0.f, 0.f, 0.f);
    for (int s = 0; s < NSEG_; ++s) {
        const float4 p = ((const float4*)(partial + ((size_t)s * B_ + b) * H_))[h4];
        acc.x += p.x; acc.y += p.y; acc.z += p.z; acc.w += p.w;
    }
    float4* P4 = (float4*)(P + (size_t)b * (2 * H_));
    P4[h4] = acc;
    P4[H_ / 4 + h4] =
        ((const float4*)(hidden + ((size_t)b * T_ + (T_ - 1)) * H_))[h4];
}

// ---------------------------------------------------------------------------
// Kernel 4 (fallback, small workspace): single-stage context + assemble P.
// ---------------------------------------------------------------------------
__global__ void k_context_fallback(const float* __restrict__ hidden,
                                   const float* __restrict__ w,
                                   float* __restrict__ P) {
    __shared__ float ws[256];
    const int b = (int)(blockIdx.x >> 2);
    const int h = ((int)(blockIdx.x & 3u) << 8) + (int)threadIdx.x;
    const float* hb = hidden + (size_t)b * T_ * H_;
    const float* wb = w + (size_t)b * T_;

    float acc = 0.f;
    for (int t0 = 0; t0 < T_; t0 += 256) {
        ws[threadIdx.x] = wb[t0 + threadIdx.x];
        __syncthreads();
        #pragma unroll 8
        for (int tt = 0; tt < 256; ++tt)
            acc = fmaf(ws[tt], hb[(size_t)(t0 + tt) * H_ + h], acc);
        __syncthreads();
    }
    P[(size_t)b * (2 * H_) + h]      = acc;
    P[(size_t)b * (2 * H_) + H_ + h] = hb[(size_t)(T_ - 1) * H_ + h];
}

// ---------------------------------------------------------------------------
// Kernel 5: out = tanh(P(32x2048) @ W_out(2048x128)) via WMMA f32 16x16x4.
//   2 M-tiles x 8 N-tiles = 16 waves = one 512-thread block.
// ---------------------------------------------------------------------------
__global__ void k_out_wmma(const float* __restrict__ P,
                           const float* __restrict__ Wout,
                           float* __restrict__ out) {
    const unsigned lane = threadIdx.x & 31u;
    const unsigned wid  = threadIdx.x >> 5;   // 0..15
    const int m0 = (int)(wid >> 3) << 4;      // 0 or 16
    const int n0 = (int)(wid & 7u) << 4;      // 0..112
    const int K  = 2 * H_;

    const int mrow = m0 + (int)(lane & 15u);
    const int ncol = n0 + (int)(lane & 15u);
    const int kh   = ((int)(lane >> 4)) << 1;

    v8f acc = {};
    for (int k = 0; k < K; k += 4) {
        v2f a, b;
        a[0] = P[(size_t)mrow * K + (k + kh)];
        a[1] = P[(size_t)mrow * K + (k + kh + 1)];
        b[0] = Wout[(size_t)(k + kh)     * DOUT_ + ncol];
        b[1] = Wout[(size_t)(k + kh + 1) * DOUT_ + ncol];
        acc = __builtin_amdgcn_wmma_f32_16x16x4_f32(
                  false, a, false, b, (short)0, acc, false, false);
    }
    const int n  = n0 + (int)(lane & 15u);
    const int mb = m0 + (((int)(lane >> 4)) << 3);
    for (int r = 0; r < 8; ++r)
        out[(size_t)(mb + r) * DOUT_ + n] = tanhf(acc[r]);
}

// ---------------------------------------------------------------------------
extern "C" void kernel_launch(void* const* d_in, const int* in_sizes, int n_in,
                              void* d_out, int out_size, void* d_ws, size_t ws_size,
                              hipStream_t stream) {
    const float* hidden = (const float*)d_in[0];   // B*T*H
    const float* Wscore = (const float*)d_in[1];   // H*H
    const float* Wout   = (const float*)d_in[2];   // 2H*DOUT
    float* out = (float*)d_out;                    // B*DOUT

    float* ws      = (float*)d_ws;
    float* v       = ws;                           // B*H   = 32768 floats
    float* score   = v + (size_t)B_ * H_;          // B*T   = 65536 floats
    float* P       = score + (size_t)B_ * T_;      // B*2H  = 65536 floats
    float* partial = P + (size_t)B_ * 2 * H_;      // NSEG*B*H = 524288 floats

    const size_t need_full =
        ((size_t)B_ * H_ + (size_t)B_ * T_ + (size_t)B_ * 2 * H_ +
         (size_t)NSEG_ * B_ * H_) * sizeof(float);

    // 1) v = h_t @ W_score   (128 waves -> 32 blocks x 128 threads)
    k_scorevec_wmma<<<32, 128, 0, stream>>>(hidden, Wscore, v);

    // 2) score[b,t] = hidden[b,t,:] . v[b,:]   (B*T waves, 8 waves/block)
    k_score<<<(B_ * T_) / 8, 256, 0, stream>>>(hidden, v, score);

    // 3) softmax over t per b
    k_softmax<<<B_, 256, 0, stream>>>(score);

    // 4) context + assemble pre_activation (wide-load two-stage if ws allows)
    if (ws_size >= need_full) {
        k_context_part<<<B_ * NSEG_, 256, 0, stream>>>(hidden, score, partial);
        k_context_reduce<<<B_, 256, 0, stream>>>(partial, hidden, P);
    } else {
        k_context_fallback<<<B_ * 4, 256, 0, stream>>>(hidden, score, P);
    }

    // 5) out = tanh(P @ W_out)   (16 waves, one block)
    k_out_wmma<<<1, 512, 0, stream>>>(P, Wout, out);
}